// vitEncoder_34600256537216
// MI455X (gfx1250) — compile-verified
//
#include <hip/hip_runtime.h>
#include <math.h>

// Problem constants (match reference)
#define NB    8
#define CC    256
#define NHD   8
#define HDIM  64
#define INNER 512
#define HID   512
#define NTOK  4096            // H*W
#define MTOT  (NB * NTOK)     // 32768 tokens

typedef float v2f __attribute__((ext_vector_type(2)));
typedef float v8f __attribute__((ext_vector_type(8)));

static __device__ __forceinline__ v8f wmma4(v2f a, v2f b, v8f c) {
  // D = A(16x4,f32) * B(4x16,f32) + C(16x16,f32)
  return __builtin_amdgcn_wmma_f32_16x16x4_f32(false, a, false, b, (short)0, c,
                                               false, false);
}

// Shared 16x16 output-tile inner loop: A row-major (lda), B row-major (ldb).
// A must point at (row m0, k=0); Bw must point at (k=0, col n0).
static __device__ __forceinline__ v8f
wmma_tile(const float* __restrict__ A, int lda, const float* __restrict__ Bw,
          int ldb, int K, int lane) {
  v8f acc = {};
  const int m  = lane & 15;          // A row / B col within tile
  const int ko = (lane >> 4) << 1;   // k sub-offset (0 or 2)
  for (int k = 0; k < K; k += 4) {
    v2f a, b;
    a.x = A[(size_t)m * lda + k + ko];
    a.y = A[(size_t)m * lda + k + ko + 1];
    b.x = Bw[(size_t)(k + ko) * ldb + m];
    b.y = Bw[(size_t)(k + ko + 1) * ldb + m];
    acc = wmma4(a, b, acc);
  }
  return acc;
}

// ---------------- LayerNorm: one wave32 per token, C=256 ----------------
__global__ void ln_kernel(const float* __restrict__ x, const float* __restrict__ g,
                          const float* __restrict__ bb, float* __restrict__ out) {
  int wave = (blockIdx.x * blockDim.x + threadIdx.x) >> 5;
  int lane = threadIdx.x & 31;
  const float* row = x + (size_t)wave * CC;
  float v[8];
  float s = 0.f;
#pragma unroll
  for (int i = 0; i < 8; ++i) { v[i] = row[lane + i * 32]; s += v[i]; }
#pragma unroll
  for (int o = 16; o; o >>= 1) s += __shfl_xor(s, o, 32);
  float mean = s * (1.0f / CC);
  float ss = 0.f;
#pragma unroll
  for (int i = 0; i < 8; ++i) { float d = v[i] - mean; ss += d * d; }
#pragma unroll
  for (int o = 16; o; o >>= 1) ss += __shfl_xor(ss, o, 32);
  float inv = rsqrtf(ss * (1.0f / CC) + 1e-6f);
  float* orow = out + (size_t)wave * CC;
#pragma unroll
  for (int i = 0; i < 8; ++i) {
    int c = lane + i * 32;
    orow[c] = (v[i] - mean) * inv * g[c] + bb[c];
  }
}

// -------- spatial max/sum partials: grid (B, 32), 128 tokens/block --------
__global__ void pool_kernel(const float* __restrict__ f, float* __restrict__ pmax,
                            float* __restrict__ psum) {
  int b = blockIdx.x, chunk = blockIdx.y, c = threadIdx.x;
  const float* base = f + ((size_t)b * NTOK + (size_t)chunk * 128) * CC + c;
  float mx = -3.4e38f, sm = 0.f;
  for (int n = 0; n < 128; ++n) {
    float v = base[(size_t)n * CC];
    mx = fmaxf(mx, v);
    sm += v;
  }
  pmax[((size_t)b * 32 + chunk) * CC + c] = mx;
  psum[((size_t)b * 32 + chunk) * CC + c] = sm;
}

// -------- CBAM channel gate: one block per batch (tiny MLP) --------
__global__ void gate_kernel(const float* __restrict__ pmax, const float* __restrict__ psum,
                            const float* __restrict__ w1, const float* __restrict__ b1,
                            const float* __restrict__ w2, const float* __restrict__ b2,
                            float* __restrict__ gate) {
  __shared__ float pm[CC], pa[CC], hm[HID], ha[HID];
  int b = blockIdx.x, t = threadIdx.x;
  float mx = -3.4e38f, sm = 0.f;
  for (int k = 0; k < 32; ++k) {
    mx = fmaxf(mx, pmax[((size_t)b * 32 + k) * CC + t]);
    sm += psum[((size_t)b * 32 + k) * CC + t];
  }
  pm[t] = mx;
  pa[t] = sm * (1.0f / NTOK);
  __syncthreads();
  for (int j = t; j < HID; j += 256) {
    float am = b1[j], aa = b1[j];
    for (int c = 0; c < CC; ++c) {
      float w = w1[(size_t)c * HID + j];
      am += pm[c] * w;
      aa += pa[c] * w;
    }
    hm[j] = fmaxf(am, 0.f);
    ha[j] = fmaxf(aa, 0.f);
  }
  __syncthreads();
  float om = b2[t], oa = b2[t];
  for (int j = 0; j < HID; ++j) {
    float w = w2[(size_t)j * CC + t];
    om += hm[j] * w;
    oa += ha[j] * w;
  }
  float s = fmaxf(om, 0.f) + fmaxf(oa, 0.f);   // relu on each MLP, then sum
  gate[(size_t)b * CC + t] = 1.0f / (1.0f + expf(-s));
}

// -------- plain GEMM Out[M,Nc] = A[M,K] @ W[K,Nc]; grid (M/16, Nc/128) --------
__global__ void gemm_plain_kernel(const float* __restrict__ A, const float* __restrict__ W,
                                  float* __restrict__ Out, int K, int Nc) {
  int lane = threadIdx.x & 31, w = threadIdx.x >> 5;
  int m0 = blockIdx.x * 16;
  int n0 = (blockIdx.y * 8 + w) * 16;
  v8f acc = wmma_tile(A + (size_t)m0 * K, K, W + n0, Nc, K, lane);
  int n = n0 + (lane & 15), hi = (lane >> 4) * 8;
#pragma unroll
  for (int j = 0; j < 8; ++j) Out[(size_t)(m0 + hi + j) * Nc + n] = acc[j];
}

// -------- per-(b,col) inverse L2 norm over N tokens; grid (B, INNER/256) ----
__global__ void rstd_kernel(const float* __restrict__ Buf, float* __restrict__ Rstd) {
  int b = blockIdx.x;
  int col = blockIdx.y * 256 + threadIdx.x;
  const float* p = Buf + (size_t)b * NTOK * INNER + col;
  float ss = 0.f;
  for (int n = 0; n < NTOK; ++n) {
    float v = p[(size_t)n * INNER];
    ss += v * v;
  }
  Rstd[(size_t)b * INNER + col] = rsqrtf(fmaxf(ss, 1e-12f));
}

// -------- attn logits: per (b,h) 64x64 = K^T-ish GEMM over N=4096 --------
// grid (B*NH, 2), 8 waves/block; wave handles one 16x16 tile (tm,tn)
__global__ void attn_kernel(const float* __restrict__ Qb, const float* __restrict__ Kb,
                            float* __restrict__ Attn) {
  int lane = threadIdx.x & 31, w = threadIdx.x >> 5;
  int bh = blockIdx.x;
  int t = blockIdx.y * 8 + w;             // 0..15
  int d0 = (t >> 2) * 16, e0 = (t & 3) * 16;
  size_t base = (size_t)(bh >> 3) * NTOK * INNER + (size_t)(bh & 7) * HDIM;
  int m = lane & 15, ko = (lane >> 4) << 1;
  v8f acc = {};
  for (int k = 0; k < NTOK; k += 4) {
    v2f a, b;
    a.x = Kb[base + (size_t)(k + ko) * INNER + d0 + m];
    a.y = Kb[base + (size_t)(k + ko + 1) * INNER + d0 + m];
    b.x = Qb[base + (size_t)(k + ko) * INNER + e0 + m];
    b.y = Qb[base + (size_t)(k + ko + 1) * INNER + e0 + m];
    acc = wmma4(a, b, acc);
  }
  int e = e0 + m, hi = (lane >> 4) * 8;
#pragma unroll
  for (int j = 0; j < 8; ++j)
    Attn[((size_t)bh * 64 + d0 + hi + j) * 64 + e] = acc[j];
}

// -------- scale by rstd_k*rstd_q then softmax over e; thread per (b,h,d) ----
__global__ void softmax_kernel(float* __restrict__ Attn, const float* __restrict__ rstdQ,
                               const float* __restrict__ rstdK) {
  int t = blockIdx.x * 256 + threadIdx.x;  // 0..4095
  int bh = t >> 6, d = t & 63;
  int b = bh >> 3, h = bh & 7;
  float* row = Attn + (size_t)t * 64;
  float rk = rstdK[(size_t)b * INNER + h * HDIM + d];
  const float* rq = rstdQ + (size_t)b * INNER + h * HDIM;
  float s[64];
  float mx = -3.4e38f;
#pragma unroll
  for (int e = 0; e < 64; ++e) { s[e] = row[e] * rk * rq[e]; mx = fmaxf(mx, s[e]); }
  float sum = 0.f;
#pragma unroll
  for (int e = 0; e < 64; ++e) { s[e] = expf(s[e] - mx); sum += s[e]; }
  float inv = 1.0f / sum;
#pragma unroll
  for (int e = 0; e < 64; ++e) row[e] = s[e] * inv;
}

// -------- xs = attn @ V, stored directly in the raw-TF-reshape layout ------
// xr[b, n', c'] with n' = h*512 + n/8 , c' = (n%8)*64 + d
// grid (B*NH, 128), 8 waves/block; wave tile: d0 in {0..48}, n0 in {0..4080}
__global__ void xs_kernel(const float* __restrict__ Attn, const float* __restrict__ Vb,
                          float* __restrict__ Xr) {
  int lane = threadIdx.x & 31, w = threadIdx.x >> 5;
  int bh = blockIdx.x;
  int t = blockIdx.y * 8 + w;              // 0..1023
  int d0 = (t >> 8) * 16;
  int n0 = (t & 255) * 16;
  int b = bh >> 3, h = bh & 7;
  size_t vbase = (size_t)b * NTOK * INNER + (size_t)h * HDIM;
  int m = lane & 15, ko = (lane >> 4) << 1;
  v8f acc = {};
#pragma unroll
  for (int k = 0; k < HDIM; k += 4) {      // K dim = e
    v2f a, bv;
    a.x  = Attn[((size_t)bh * 64 + d0 + m) * 64 + k + ko];
    a.y  = Attn[((size_t)bh * 64 + d0 + m) * 64 + k + ko + 1];
    bv.x = Vb[vbase + (size_t)(n0 + m) * INNER + k + ko];
    bv.y = Vb[vbase + (size_t)(n0 + m) * INNER + k + ko + 1];
    acc = wmma4(a, bv, acc);
  }
  int n = n0 + m, hi = (lane >> 4) * 8;
  size_t outbase = ((size_t)b * NTOK + (size_t)h * 512 + (n >> 3)) * INNER +
                   (size_t)(n & 7) * 64 + d0 + hi;
#pragma unroll
  for (int j = 0; j < 8; ++j) Xr[outbase + j] = acc[j];
}

// -------- x3 = (xr @ wo) * gate + f ; grid (M/16, 2) --------
__global__ void wo_kernel(const float* __restrict__ Xr, const float* __restrict__ Wo,
                          const float* __restrict__ gate, const float* __restrict__ Fln,
                          float* __restrict__ X3) {
  int lane = threadIdx.x & 31, w = threadIdx.x >> 5;
  int m0 = blockIdx.x * 16;
  int n0 = (blockIdx.y * 8 + w) * 16;
  v8f acc = wmma_tile(Xr + (size_t)m0 * INNER, INNER, Wo + n0, CC, INNER, lane);
  int n = n0 + (lane & 15), hi = (lane >> 4) * 8;
#pragma unroll
  for (int j = 0; j < 8; ++j) {
    int mrow = m0 + hi + j;
    int b = mrow >> 12;
    X3[(size_t)mrow * CC + n] =
        acc[j] * gate[(size_t)b * CC + n] + Fln[(size_t)mrow * CC + n];
  }
}

// -------- hidden = relu(y @ w1 + b1) ; grid (M/16, 4) --------
__global__ void mlp1_kernel(const float* __restrict__ Y, const float* __restrict__ W1,
                            const float* __restrict__ B1, float* __restrict__ Hid) {
  int lane = threadIdx.x & 31, w = threadIdx.x >> 5;
  int m0 = blockIdx.x * 16;
  int n0 = (blockIdx.y * 8 + w) * 16;
  v8f acc = wmma_tile(Y + (size_t)m0 * CC, CC, W1 + n0, HID, CC, lane);
  int n = n0 + (lane & 15), hi = (lane >> 4) * 8;
  float bia = B1[n];
#pragma unroll
  for (int j = 0; j < 8; ++j)
    Hid[(size_t)(m0 + hi + j) * HID + n] = fmaxf(acc[j] + bia, 0.f);
}

// -------- out = x3 + relu(hidden @ w2 + b2) ; grid (M/16, 2) --------
__global__ void mlp2_kernel(const float* __restrict__ Hid, const float* __restrict__ W2,
                            const float* __restrict__ B2, const float* __restrict__ X3,
                            float* __restrict__ Out) {
  int lane = threadIdx.x & 31, w = threadIdx.x >> 5;
  int m0 = blockIdx.x * 16;
  int n0 = (blockIdx.y * 8 + w) * 16;
  v8f acc = wmma_tile(Hid + (size_t)m0 * HID, HID, W2 + n0, CC, HID, lane);
  int n = n0 + (lane & 15), hi = (lane >> 4) * 8;
  float bia = B2[n];
#pragma unroll
  for (int j = 0; j < 8; ++j) {
    size_t idx = (size_t)(m0 + hi + j) * CC + n;
    Out[idx] = X3[idx] + fmaxf(acc[j] + bia, 0.f);
  }
}

extern "C" void kernel_launch(void* const* d_in, const int* in_sizes, int n_in,
                              void* d_out, int out_size, void* d_ws, size_t ws_size,
                              hipStream_t stream) {
  const float* x     = (const float*)d_in[0];
  const float* ln1_g = (const float*)d_in[1];
  const float* ln1_b = (const float*)d_in[2];
  const float* wq    = (const float*)d_in[3];
  const float* wk    = (const float*)d_in[4];
  const float* wv    = (const float*)d_in[5];
  const float* wo    = (const float*)d_in[6];
  const float* cb_w1 = (const float*)d_in[7];
  const float* cb_b1 = (const float*)d_in[8];
  const float* cb_w2 = (const float*)d_in[9];
  const float* cb_b2 = (const float*)d_in[10];
  const float* ln4_g = (const float*)d_in[11];
  const float* ln4_b = (const float*)d_in[12];
  const float* m_w1  = (const float*)d_in[13];
  const float* m_b1  = (const float*)d_in[14];
  const float* m_w2  = (const float*)d_in[15];
  const float* m_b2  = (const float*)d_in[16];
  float* out = (float*)d_out;

  // Workspace layout (floats), with buffer reuse:
  const size_t F  = (size_t)MTOT * CC;      // 8.39M
  const size_t QK = (size_t)MTOT * INNER;   // 16.78M
  float* ws    = (float*)d_ws;
  float* f     = ws;                 // LN1 output; reused as LN4 output (y)
  float* qbuf  = f + F;              // Q; reused as xr (attn output)
  float* kbuf  = qbuf + QK;          // K; reused as MLP hidden
  float* vbuf  = kbuf + QK;          // V; reused as x3
  float* attn  = vbuf + QK;          // [B*NH, 64, 64]
  float* rstdq = attn + (size_t)NB * NHD * 64 * 64;
  float* rstdk = rstdq + (size_t)NB * INNER;
  float* pmax  = rstdk + (size_t)NB * INNER;
  float* psum  = pmax + (size_t)NB * 32 * CC;
  float* gate  = psum + (size_t)NB * 32 * CC;
  float* xr  = qbuf;   // Q dead after attn logits
  float* x3  = vbuf;   // V dead after xs
  float* y   = f;      // f dead after wo epilogue
  float* hid = kbuf;   // K dead after attn logits

  dim3 blk(256);

  // 1) f = LN1(x)
  ln_kernel<<<dim3(MTOT / 8), blk, 0, stream>>>(x, ln1_g, ln1_b, f);
  // 2) pooling partials + gate
  pool_kernel<<<dim3(NB, 32), blk, 0, stream>>>(f, pmax, psum);
  gate_kernel<<<dim3(NB), blk, 0, stream>>>(pmax, psum, cb_w1, cb_b1, cb_w2,
                                            cb_b2, gate);
  // 3) Q, K, V token-major GEMMs (WMMA f32)
  gemm_plain_kernel<<<dim3(MTOT / 16, INNER / 128), blk, 0, stream>>>(f, wq, qbuf, CC, INNER);
  gemm_plain_kernel<<<dim3(MTOT / 16, INNER / 128), blk, 0, stream>>>(f, wk, kbuf, CC, INNER);
  gemm_plain_kernel<<<dim3(MTOT / 16, INNER / 128), blk, 0, stream>>>(f, wv, vbuf, CC, INNER);
  // 4) L2-norm factors for q, k (per b, per column)
  rstd_kernel<<<dim3(NB, INNER / 256), blk, 0, stream>>>(qbuf, rstdq);
  rstd_kernel<<<dim3(NB, INNER / 256), blk, 0, stream>>>(kbuf, rstdk);
  // 5) channel-attention logits, scaled softmax, attn@V into raw-reshape xr
  attn_kernel<<<dim3(NB * NHD, 2), blk, 0, stream>>>(qbuf, kbuf, attn);
  softmax_kernel<<<dim3(NB * NHD * 64 / 256), blk, 0, stream>>>(attn, rstdq, rstdk);
  xs_kernel<<<dim3(NB * NHD, 128), blk, 0, stream>>>(attn, vbuf, xr);
  // 6) x3 = (xr@wo)*gate + f ; out = x3 + MLP(LN4(x3))
  wo_kernel<<<dim3(MTOT / 16, CC / 128), blk, 0, stream>>>(xr, wo, gate, f, x3);
  ln_kernel<<<dim3(MTOT / 8), blk, 0, stream>>>(x3, ln4_g, ln4_b, y);
  mlp1_kernel<<<dim3(MTOT / 16, HID / 128), blk, 0, stream>>>(y, m_w1, m_b1, hid);
  mlp2_kernel<<<dim3(MTOT / 16, CC / 128), blk, 0, stream>>>(hid, m_w2, m_b2, x3, out);
}